// VariationalQuantumCircuit_83382495084753
// MI455X (gfx1250) — compile-verified
//
#include <hip/hip_runtime.h>
#include <math.h>

// ---------------------------------------------------------------------------
// VariationalQuantumCircuit on MI455X (gfx1250, wave32, WMMA).
//
// out[32768, 512] = state[32768, 512] @ T[512, 512]
// with T = (Trot0@Tent0)(Trot1@Tent1)(Trot2@Tent2)(Trot3@Tent3),
// all fp32 to match the reference, using V_WMMA_F32_16X16X4_F32.
// ---------------------------------------------------------------------------

typedef float v2f __attribute__((ext_vector_type(2)));
typedef float v4f __attribute__((ext_vector_type(4)));
typedef float v8f __attribute__((ext_vector_type(8)));

#define DQ      512     // state dim D
#define QQ      256     // qubits
#define NLAY    4
#define NPAIR   255
#define KSTEPS  128     // D / 4   (K steps of the 16x16x4 WMMA)
#define NTILES  32      // D / 16  (N tiles)
#define M_STATE (16 * 2048)

static __device__ __forceinline__ v8f wmma_f32_4(v2f a, v2f b, v8f c) {
  // D = A(16x4) * B(4x16) + C(16x16), fp32
  return __builtin_amdgcn_wmma_f32_16x16x4_f32(false, a, false, b, (short)0, c,
                                               false, false);
}

// ---------------------------------------------------------------------------
// T_ent closed form. Scan step p does (on every row v of the carry):
//   v[p+1] += c_p * v[p];  v[p] += c_p * v[p+1]_old
// Row r result: T[r][r]=1; T[r][r-1]=c_{r-1}; T[r][r+k]=c_r*...*c_{r+k-1}
// (only indices 0..255 participate; rows>=256 are identity).
// grid: (NLAY) blocks x 512 threads; thread r owns row r.
// ---------------------------------------------------------------------------
__global__ void vqc_build_ent(const float* __restrict__ ent_params,
                              float* __restrict__ Tent) {
  const int l = blockIdx.x;
  const int r = threadIdx.x;

  __shared__ float c[NPAIR];
  for (int p = threadIdx.x; p < NPAIR; p += blockDim.x) {
    float x = ent_params[l * NPAIR + p];
    c[p] = 0.1f * (1.0f / (1.0f + expf(-x)));   // sigmoid(x)*0.1
  }
  __syncthreads();

  float* row = Tent + ((size_t)l * DQ + r) * DQ;
  for (int j = 0; j < DQ; ++j) row[j] = 0.0f;
  row[r] = 1.0f;
  if (r < QQ) {
    if (r >= 1) row[r - 1] = c[r - 1];
    if (r <= QQ - 2) {
      float val = c[r];
      for (int j = r + 1; j < QQ; ++j) {
        row[j] = val;
        if (j < NPAIR) val *= c[j];
      }
    }
  }
}

// ---------------------------------------------------------------------------
// T_layer[l] = T_rot[l] @ T_ent[l].
// T_rot is block-diagonal with per-qubit 2x2 block  Rx@Ry@Rz (real parts):
//   a = c0, block = [[ (a*c1)*c2, (a*(-s1))*c2 ], [ (a*s1)*c2, (a*c1)*c2 ]]
// Row i = 2q+par of T_layer is a 2-term combo of T_ent rows 2q, 2q+1.
// grid: (512, 4) blocks x 128 threads.
// ---------------------------------------------------------------------------
__global__ void vqc_build_layer(const float* __restrict__ rot_params,
                                const float* __restrict__ Tent,
                                float* __restrict__ Tlay) {
  const int i = blockIdx.x;      // row 0..511
  const int l = blockIdx.y;      // layer
  const int q = i >> 1;
  const int par = i & 1;

  const float t0 = rot_params[((l * QQ) + q) * 3 + 0];
  const float t1 = rot_params[((l * QQ) + q) * 3 + 1];
  const float t2 = rot_params[((l * QQ) + q) * 3 + 2];
  const float c0 = cosf(t0 * 0.5f);
  const float c1 = cosf(t1 * 0.5f);
  const float s1 = sinf(t1 * 0.5f);
  const float c2 = cosf(t2 * 0.5f);
  const float b00 = (c0 * c1) * c2;      // == b11
  const float b01 = (c0 * (-s1)) * c2;
  const float b10 = (c0 * s1) * c2;

  const float w0 = (par == 0) ? b00 : b10;  // coeff of T_ent row 2q
  const float w1 = (par == 0) ? b01 : b00;  // coeff of T_ent row 2q+1

  const float* r0 = Tent + ((size_t)l * DQ + 2 * q) * DQ;
  const float* r1 = r0 + DQ;
  float* out = Tlay + ((size_t)l * DQ + i) * DQ;
  for (int j = threadIdx.x; j < DQ; j += blockDim.x)
    out[j] = w0 * r0[j] + w1 * r1[j];
}

// ---------------------------------------------------------------------------
// Pack a row-major B[512][512] into WMMA 16x16x4 B-fragment order:
//   Bp[ks][nt][lane] = { B[k0][col], B[k0+1][col] }
//   k0 = 4*ks + 2*(lane>=16), col = 16*nt + (lane&15)
// so a wave's B-fragment load is one fully-coalesced 256B global_load_b64.
// grid: 512 blocks x 256 threads (128*32*32 fragments-lanes).
// ---------------------------------------------------------------------------
__global__ void vqc_pack_b(const float* __restrict__ B, float* __restrict__ Bp) {
  const int idx = blockIdx.x * blockDim.x + threadIdx.x;   // 0..131071
  const int lane = idx & 31;
  const int nt = (idx >> 5) & 31;
  const int ks = idx >> 10;
  const int k0 = ks * 4 + ((lane >> 4) << 1);
  const int col = nt * 16 + (lane & 15);
  v2f v;
  v[0] = B[(size_t)k0 * DQ + col];
  v[1] = B[(size_t)(k0 + 1) * DQ + col];
  *(v2f*)(Bp + (size_t)idx * 2) = v;
}

// ---------------------------------------------------------------------------
// C[M,512] = A[M,512] @ B[512,512]   (B pre-packed by vqc_pack_b)
// Block = 128 threads (4 wave32), owns 32 rows x 512 cols.
// A tile staged in LDS in *fragment order* -> conflict-free ds_load_b64.
// Wave w covers N tiles [8w, 8w+8); 2 M-tiles x 8 N-tiles = 16 v8f accums.
// grid: M/32 blocks.
// ---------------------------------------------------------------------------
__global__ void __launch_bounds__(128)
vqc_wmma_gemm(const float* __restrict__ A, const float* __restrict__ Bp,
              float* __restrict__ C) {
  __shared__ v2f As[2 * KSTEPS * 32];   // 64 KB: [mtile][ks][lane] fragments

  const int row0 = blockIdx.x * 32;

  // ---- stage A[row0..row0+32) x [0..512) into fragment-ordered LDS ----
  {
    const int row = threadIdx.x >> 2;        // 0..31
    const int cq = threadIdx.x & 3;          // 4 threads per row
    const int mt = row >> 4;
    const int lr = row & 15;
    const float* arow = A + (size_t)(row0 + row) * DQ;
#pragma unroll 4
    for (int s = 0; s < 32; ++s) {
      const int ks = cq + 4 * s;             // K chunk [4ks, 4ks+4)
      v4f v = *(const v4f*)(arow + ks * 4);
      v2f lo; lo[0] = v[0]; lo[1] = v[1];    // lanes 0..15 : K = 4ks, 4ks+1
      v2f hi; hi[0] = v[2]; hi[1] = v[3];    // lanes 16..31: K = 4ks+2, 4ks+3
      As[(mt * KSTEPS + ks) * 32 + lr] = lo;
      As[(mt * KSTEPS + ks) * 32 + lr + 16] = hi;
    }
  }
  __syncthreads();

  const int wave = threadIdx.x >> 5;
  const int lane = threadIdx.x & 31;
  const int ntbase = wave * 8;

  v8f acc[2][8];
#pragma unroll
  for (int m = 0; m < 2; ++m)
#pragma unroll
    for (int t = 0; t < 8; ++t)
      acc[m][t] = (v8f){0.f, 0.f, 0.f, 0.f, 0.f, 0.f, 0.f, 0.f};

  const v2f* BpV = (const v2f*)Bp;
  for (int ks = 0; ks < KSTEPS; ++ks) {
    const v2f a0 = As[(0 * KSTEPS + ks) * 32 + lane];
    const v2f a1 = As[(1 * KSTEPS + ks) * 32 + lane];
#pragma unroll
    for (int t = 0; t < 8; ++t) {
      const v2f b = BpV[((size_t)ks * NTILES + (ntbase + t)) * 32 + lane];
      acc[0][t] = wmma_f32_4(a0, b, acc[0][t]);
      acc[1][t] = wmma_f32_4(a1, b, acc[1][t]);
    }
  }

  // ---- store D: VGPR v, lanes 0-15 -> (M=v, N=lane); 16-31 -> (M=8+v) ----
  const int lrow = lane & 15;
  const int hi8 = (lane >> 4) * 8;
#pragma unroll
  for (int mt = 0; mt < 2; ++mt)
#pragma unroll
    for (int t = 0; t < 8; ++t) {
      const int col = (ntbase + t) * 16 + lrow;
#pragma unroll
      for (int v = 0; v < 8; ++v) {
        const int r = row0 + mt * 16 + hi8 + v;
        C[(size_t)r * DQ + col] = acc[mt][t][v];
      }
    }
}

// ---------------------------------------------------------------------------
// Host-side launch. Workspace layout (floats), total 8 MB:
//   [0       , 1048576) : T_ent (4x512x512), later reused:
//        T01 @ 0, T23 @ 262144, Ttot @ 524288, Bp @ 786432
//   [1048576 , 2097152) : T_layer (4x512x512)
// ---------------------------------------------------------------------------
extern "C" void kernel_launch(void* const* d_in, const int* in_sizes, int n_in,
                              void* d_out, int out_size, void* d_ws,
                              size_t ws_size, hipStream_t stream) {
  (void)in_sizes; (void)n_in; (void)out_size; (void)ws_size;
  const float* state = (const float*)d_in[0];  // [16,2048,512]
  const float* rotp  = (const float*)d_in[1];  // [4,256,3]
  const float* entp  = (const float*)d_in[2];  // [4,255]
  float* out = (float*)d_out;                  // [16,2048,512]

  float* w     = (float*)d_ws;
  float* Tent  = w;                 // 4 MB (dead after build_layer)
  float* Tlay  = w + 1048576;       // 4 MB
  float* T01   = w + 0;
  float* T23   = w + 262144;
  float* Ttot  = w + 524288;
  float* Bp    = w + 786432;        // 1 MB packed-B scratch

  // 1) entanglement transforms (closed form of the scan)
  vqc_build_ent<<<NLAY, 512, 0, stream>>>(entp, Tent);

  // 2) T_layer[l] = T_rot[l] @ T_ent[l]
  vqc_build_layer<<<dim3(DQ, NLAY), 128, 0, stream>>>(rotp, Tent, Tlay);

  // 3) T01 = T_layer0 @ T_layer1
  vqc_pack_b<<<512, 256, 0, stream>>>(Tlay + 1 * 262144, Bp);
  vqc_wmma_gemm<<<DQ / 32, 128, 0, stream>>>(Tlay + 0 * 262144, Bp, T01);

  // 4) T23 = T_layer2 @ T_layer3
  vqc_pack_b<<<512, 256, 0, stream>>>(Tlay + 3 * 262144, Bp);
  vqc_wmma_gemm<<<DQ / 32, 128, 0, stream>>>(Tlay + 2 * 262144, Bp, T23);

  // 5) Ttot = T01 @ T23
  vqc_pack_b<<<512, 256, 0, stream>>>(T23, Bp);
  vqc_wmma_gemm<<<DQ / 32, 128, 0, stream>>>(T01, Bp, Ttot);

  // 6) out = state @ Ttot   (M = 32768)
  vqc_pack_b<<<512, 256, 0, stream>>>(Ttot, Bp);
  vqc_wmma_gemm<<<M_STATE / 32, 128, 0, stream>>>(state, Bp, out);
}